// Chamfer_9749575762307
// MI455X (gfx1250) — compile-verified
//
#include <hip/hip_runtime.h>

typedef __attribute__((ext_vector_type(2))) float v2f;
typedef __attribute__((ext_vector_type(8))) float v8f;

#define CHUNK 2048            // candidate points staged per LDS refill
#define QPAD  16              // 16 float2 = 128 B = 32-bank shift between halves

// One nearest-neighbor pass: for each row-point in P, find argmin / min of
// squared distance over all col-points in Q.
//   P: [B, Nr, 3]  (query rows, 16 per wave)
//   Q: [B, Mc, 3]  (candidates, streamed through LDS)
// WMMA formulation: A = [x,y,z,1], B = [-2qx,-2qy,-2qz,|q|^2]
//   => D = |q|^2 - 2 p.q    (true dist = D + |p|^2, added at the end)
__global__ __launch_bounds__(256)
void chamfer_nn_pass(const float* __restrict__ P,
                     const float* __restrict__ Q,
                     int Nr, int Mc,
                     int*   __restrict__ idx_out,    // [B*Nr]
                     float* __restrict__ dist_out)   // [B*Nr]
{
    // half-split candidate staging:
    //   qs2[i]            = (-2x, -2y)        (K components 0,1 -> lanes 0..15)
    //   qs2[CHUNK+QPAD+i] = (-2z, |q|^2)      (K components 2,3 -> lanes 16..31)
    __shared__ float2 qs2[2 * CHUNK + QPAD];

    const int tid  = threadIdx.x;
    const int lane = tid & 31;
    const int wave = tid >> 5;
    const int lo   = lane & 15;   // column / row-within-tile index
    const int hi   = lane >> 4;   // which half of K / which row half

    const int rowsPerWG  = 128;               // 8 waves * 16 rows
    const int wgPerBatch = Nr / rowsPerWG;
    const int b    = blockIdx.x / wgPerBatch;
    const int nblk = blockIdx.x % wgPerBatch;
    const int n0   = nblk * rowsPerWG + wave * 16;  // first row of this wave

    // --- loop-invariant A matrix (16x4 f32): lane holds row (n0+lo),
    //     K components (2*hi, 2*hi+1) of [x, y, z, 1]
    const float* prow = P + ((size_t)b * Nr + (size_t)(n0 + lo)) * 3;
    const float px = prow[0], py = prow[1], pz = prow[2];
    v2f a;
    a.x = hi ? pz   : px;
    a.y = hi ? 1.0f : py;

    float best[8];
    int   bidx[8];
#pragma unroll
    for (int v = 0; v < 8; ++v) { best[v] = 3.4e38f; bidx[v] = 0; }

    const float* qb   = Q + (size_t)b * Mc * 3;
    const int    qoff = hi * (CHUNK + QPAD) + lo;   // per-lane LDS read base

    for (int cb = 0; cb < Mc; cb += CHUNK) {
        __syncthreads();   // previous chunk fully consumed
        // stage + pre-transform candidates
        for (int i = tid; i < CHUNK; i += 256) {
            const float* q = qb + (size_t)(cb + i) * 3;
            const float x = q[0], y = q[1], z = q[2];
            qs2[i]                = make_float2(-2.0f * x, -2.0f * y);
            qs2[CHUNK + QPAD + i] = make_float2(-2.0f * z, x * x + y * y + z * z);
        }
        __syncthreads();

#pragma unroll 4
        for (int mt = 0; mt < CHUNK / 16; ++mt) {
            const float2 q2 = qs2[qoff + mt * 16];
            v2f bb;
            bb.x = q2.x;
            bb.y = q2.y;

            v8f c = {};
            c = __builtin_amdgcn_wmma_f32_16x16x4_f32(
                    /*neg_a=*/false, a, /*neg_b=*/false, bb,
                    /*c_mod=*/(short)0, c, /*reuse_a=*/false, /*reuse_b=*/false);

            const int mcand = cb + mt * 16 + lo;   // this lane's column id
#pragma unroll
            for (int v = 0; v < 8; ++v) {
                // strict < keeps the earliest (smallest) m per column class
                if (c[v] < best[v]) { best[v] = c[v]; bidx[v] = mcand; }
            }
        }
    }

    // --- cross-lane argmin over the 16 column classes (stays within each
    //     16-lane half; rows 0..7 live in lanes 0..15, rows 8..15 in 16..31)
#pragma unroll
    for (int off = 1; off < 16; off <<= 1) {
#pragma unroll
        for (int v = 0; v < 8; ++v) {
            const float ov = __shfl_xor(best[v], off, 32);
            const int   oi = __shfl_xor(bidx[v], off, 32);
            if (ov < best[v] || (ov == best[v] && oi < bidx[v])) {
                best[v] = ov; bidx[v] = oi;
            }
        }
    }

    // --- lane lo==v in each half writes row (v + 8*hi)
#pragma unroll
    for (int v = 0; v < 8; ++v) {
        if (lo == v) {
            const int n = n0 + v + 8 * hi;
            const float* pr = P + ((size_t)b * Nr + (size_t)n) * 3;
            const float n1 = pr[0] * pr[0] + pr[1] * pr[1] + pr[2] * pr[2];
            idx_out [(size_t)b * Nr + n] = bidx[v];
            dist_out[(size_t)b * Nr + n] = best[v] + n1;
        }
    }
}

extern "C" void kernel_launch(void* const* d_in, const int* in_sizes, int n_in,
                              void* d_out, int out_size, void* d_ws, size_t ws_size,
                              hipStream_t stream) {
    const float* p1 = (const float*)d_in[0];   // [B,N,3]
    const float* p2 = (const float*)d_in[1];   // [B,M,3]

    const int B = 8;
    const int N = in_sizes[0] / (B * 3);       // 4096
    const int M = in_sizes[1] / (B * 3);       // 4096
    const size_t BN = (size_t)B * N;
    const size_t BM = (size_t)B * M;

    int*   oi = (int*)d_out;
    float* of = (float*)d_out;
    // d_out layout (flat, return order): idx1[BN] | idx2[BM] | dist1[BN] | dist2[BM]

    dim3 blk(256);

    // pass 1: rows = points1, cols = points2  -> idx1, dist1
    chamfer_nn_pass<<<dim3(B * (N / 128)), blk, 0, stream>>>(
        p1, p2, N, M, oi, of + BN + BM);

    // pass 2: rows = points2, cols = points1  -> idx2, dist2
    chamfer_nn_pass<<<dim3(B * (M / 128)), blk, 0, stream>>>(
        p2, p1, M, N, oi + BN, of + 2 * BN + BM);
}